// SimpleModel_10900626997523
// MI455X (gfx1250) — compile-verified
//
#include <hip/hip_runtime.h>
#include <math.h>

#define EPS 1e-6f
#define DIM 768

typedef __attribute__((ext_vector_type(2))) float v2f;
typedef __attribute__((ext_vector_type(8))) float v8f;

// ---------------------------------------------------------------------------
// Kernel 1: q_emb = mean(table1[q_ids], axis=0)  and  q_norm = max(||q||, eps)
// One block, 256 threads. ws layout: qe[0..767], qe[768] = q_norm.
// ---------------------------------------------------------------------------
__global__ void __launch_bounds__(256) q_embed_kernel(
    const float* __restrict__ table1, const int* __restrict__ q_ids,
    int lq, float* __restrict__ qe) {
  __shared__ float red[256];
  const int t = threadIdx.x;
  float sq = 0.0f;
  const float inv = 1.0f / (float)lq;
  for (int d = t; d < DIM; d += 256) {
    float acc = 0.0f;
    for (int i = 0; i < lq; ++i) {
      acc += table1[(long)q_ids[i] * DIM + d];
    }
    acc *= inv;
    qe[d] = acc;
    sq += acc * acc;
  }
  red[t] = sq;
  __syncthreads();
  for (int s = 128; s > 0; s >>= 1) {
    if (t < s) red[t] += red[t + s];
    __syncthreads();
  }
  if (t == 0) qe[DIM] = fmaxf(sqrtf(red[0]), EPS);
}

// ---------------------------------------------------------------------------
// Kernel 2: per-segment mean embedding + norm.
// One block per segment. cand_segments is sorted -> binary search [lo,hi).
// Threads 0..191 each stream one float4 lane of the 768-float rows (coalesced);
// all 256 threads join the LDS norm reduction.
// ---------------------------------------------------------------------------
__global__ void __launch_bounds__(256) segment_mean_kernel(
    const float* __restrict__ table2, const int* __restrict__ cand_ids,
    const int* __restrict__ cand_segments, int T,
    float* __restrict__ cemb, float* __restrict__ cnorm) {
  const int s = blockIdx.x;
  const int t = threadIdx.x;
  __shared__ int s_lo, s_hi;
  if (t == 0) {
    int lo = 0, hi = T;
    while (lo < hi) { int mid = (lo + hi) >> 1; if (cand_segments[mid] < s) lo = mid + 1; else hi = mid; }
    s_lo = lo;
    int lo2 = lo, hi2 = T;
    while (lo2 < hi2) { int mid = (lo2 + hi2) >> 1; if (cand_segments[mid] < s + 1) lo2 = mid + 1; else hi2 = mid; }
    s_hi = lo2;
  }
  __syncthreads();
  const int lo = s_lo, hi = s_hi;
  const bool active = (t < 192);
  const int d4 = t * 4;

  float4 acc = make_float4(0.f, 0.f, 0.f, 0.f);
  for (int tok = lo; tok < hi; ++tok) {
    const long row = (long)cand_ids[tok];
    if (active) {
      if (tok + 1 < hi) {
        // gfx1250 global_prefetch_b8 on the next gathered row
        __builtin_prefetch(table2 + (long)cand_ids[tok + 1] * DIM + d4, 0, 1);
      }
      const float4 v = *(const float4*)(table2 + row * DIM + d4);
      acc.x += v.x; acc.y += v.y; acc.z += v.z; acc.w += v.w;
    }
  }
  const float scale = 1.0f / fmaxf((float)(hi - lo), 1.0f);
  float sq = 0.0f;
  if (active) {
    acc.x *= scale; acc.y *= scale; acc.z *= scale; acc.w *= scale;
    *(float4*)(cemb + (long)s * DIM + d4) = acc;
    sq = acc.x * acc.x + acc.y * acc.y + acc.z * acc.z + acc.w * acc.w;
  }
  __shared__ float red[256];
  red[t] = sq;
  __syncthreads();
  for (int st = 128; st > 0; st >>= 1) {
    if (t < st) red[t] += red[t + st];
    __syncthreads();
  }
  if (t == 0) cnorm[s] = fmaxf(sqrtf(red[0]), EPS);
}

// ---------------------------------------------------------------------------
// Kernel 3: cos[s0+m] = dot(cand_emb[s0+m], q) / (q_norm * c_norm[s0+m])
// One wave (32 lanes) per 16 candidates, using V_WMMA_F32_16X16X4_F32.
// A layout (ISA 7.12.2, 32-bit A 16x4): lanes 0-15 -> M=lane, K={k0,k0+1};
// lanes 16-31 -> M=lane-16, K={k0+2,k0+3}. B carries q replicated over all
// 16 columns, so every column of D holds the dot product.
// ---------------------------------------------------------------------------
__global__ void __launch_bounds__(32) cosine_wmma_kernel(
    const float* __restrict__ cemb, const float* __restrict__ cnorm,
    const float* __restrict__ qe, float* __restrict__ out) {
  const int lane = threadIdx.x;
  const int s0 = blockIdx.x * 16;
  const int m = lane & 15;
  const int ksel = (lane >> 4) << 1;  // 0 for lanes 0-15, 2 for lanes 16-31

  const float* __restrict__ arow = cemb + (long)(s0 + m) * DIM + ksel;
  const float* __restrict__ brow = qe + ksel;

  v8f c = {0.f, 0.f, 0.f, 0.f, 0.f, 0.f, 0.f, 0.f};
#pragma unroll 4
  for (int k0 = 0; k0 < DIM; k0 += 4) {
    v2f a = *(const v2f*)(arow + k0);
    v2f b = *(const v2f*)(brow + k0);
    c = __builtin_amdgcn_wmma_f32_16x16x4_f32(
        /*neg_a=*/false, a, /*neg_b=*/false, b,
        /*c_mod=*/(short)0, c, /*reuse_a=*/false, /*reuse_b=*/false);
  }

  const float qn = qe[DIM];
  // D layout: lanes 0-15 (N=lane) hold M=0..7 in c[0..7]; lanes 16-31 hold M=8..15.
  if (m == 0) {  // lanes 0 and 16 extract column N=0
    const int base = s0 + ((lane >> 4) << 3);
#pragma unroll
    for (int r = 0; r < 8; ++r) {
      out[base + r] = c[r] / (qn * cnorm[base + r]);
    }
  }
}

// ---------------------------------------------------------------------------
// Host launcher
// ---------------------------------------------------------------------------
extern "C" void kernel_launch(void* const* d_in, const int* in_sizes, int n_in,
                              void* d_out, int out_size, void* d_ws, size_t ws_size,
                              hipStream_t stream) {
  const float* table1        = (const float*)d_in[0];
  const float* table2        = (const float*)d_in[1];
  const int*   q_ids         = (const int*)d_in[2];
  const int*   cand_ids      = (const int*)d_in[3];
  const int*   cand_segments = (const int*)d_in[4];
  // d_in[5] = num_candidates scalar; N is also out_size (output is 1 x N).
  const int LQ = in_sizes[2];
  const int T  = in_sizes[3];
  const int N  = out_size;

  float* ws    = (float*)d_ws;
  float* qe    = ws;                        // [768] + q_norm at [768]
  float* cemb  = ws + 1024;                 // [N*768]
  float* cnorm = cemb + (size_t)N * DIM;    // [N]
  float* out   = (float*)d_out;

  q_embed_kernel<<<1, 256, 0, stream>>>(table1, q_ids, LQ, qe);
  segment_mean_kernel<<<N, 256, 0, stream>>>(table2, cand_ids, cand_segments, T,
                                             cemb, cnorm);
  cosine_wmma_kernel<<<N / 16, 32, 0, stream>>>(cemb, cnorm, qe, out);
}